// AlphaEstimate_27444841021647
// MI455X (gfx1250) — compile-verified
//
#include <hip/hip_runtime.h>
#include <math.h>

typedef __attribute__((ext_vector_type(2))) float v2f;
typedef __attribute__((ext_vector_type(8))) float v8f;
typedef __attribute__((ext_vector_type(4))) unsigned int v4u;
typedef __attribute__((ext_vector_type(8))) int v8i;
typedef __attribute__((ext_vector_type(4))) int v4i;

#define N_AGENTS 8
#define T_DIM    128
#define B_DIM    64
#define TB       8192      // T*B
#define S_SAMP   16
#define D_STATE  128
#define E_DIM    64
#define HH_DIM   128
#define ROWS     16        // tb rows per workgroup
#define SLDA     132       // padded LDS stride for 128-wide tiles (bank-friendly)
#define SLDB     72        // padded LDS stride for 64-wide tiles

#if defined(__has_builtin)
#if __has_builtin(__builtin_amdgcn_tensor_load_to_lds) && __has_builtin(__builtin_amdgcn_s_wait_tensorcnt)
#define USE_TDM 1
#endif
#endif
#ifndef USE_TDM
#define USE_TDM 0
#endif

__device__ __forceinline__ unsigned pcg_hash(unsigned x) {
    unsigned state = x * 747796405u + 2891336453u;
    unsigned word  = ((state >> ((state >> 28u) + 4u)) ^ state) * 277803737u;
    return (word >> 22u) ^ word;
}

// One 16x16 f32 output tile: A (16xK, f32, LDS, stride lda) x B (KxN, f32, global, stride ldb).
// V_WMMA_F32_16X16X4_F32 layout: A 16x4 -> lanes 0-15 hold K={0,1}, lanes 16-31 K={2,3};
// B 4x16 row-striped across lanes (VGPR v: K=v / K=v+2). Whole wave, EXEC all ones.
__device__ __forceinline__ v8f wmma_gemm_tile(const float* A, int lda,
                                              const float* __restrict__ B, int ldb,
                                              int c0, int K, int lane) {
    const int half = lane >> 4;
    const int lm   = lane & 15;
    v8f acc = {};
#pragma unroll 4
    for (int k0 = 0; k0 < K; k0 += 4) {
        const int ka = k0 + 2 * half;
        v2f av, bv;
        av.x = A[lm * lda + ka];
        av.y = A[lm * lda + ka + 1];
        bv.x = B[ka * ldb + c0 + lm];
        bv.y = B[(ka + 1) * ldb + c0 + lm];
        acc = __builtin_amdgcn_wmma_f32_16x16x4_f32(false, av, false, bv,
                                                    (short)0, acc, false, false);
    }
    return acc;
}

// Store a 16x16 C/D tile to LDS with fused bias + activation. act: 0=none, 1=relu, 2=abs.
__device__ __forceinline__ void store_tile(float* O, int ldo, int c0, int lane,
                                           v8f acc, const float* __restrict__ bias, int act) {
    const int half = lane >> 4;
    const int lm   = lane & 15;
    const float bv = bias[c0 + lm];
#pragma unroll
    for (int v = 0; v < 8; ++v) {
        float x = acc[v] + bv;
        if (act == 1)      x = fmaxf(x, 0.0f);
        else if (act == 2) x = fabsf(x);
        O[(v + 8 * half) * ldo + c0 + lm] = x;
    }
}

extern "C" __global__ __launch_bounds__(128)
void alpha_estimate_kernel(const float* __restrict__ q_vals, const float* __restrict__ states,
                           const float* __restrict__ hw1_k1, const float* __restrict__ hw1_b1,
                           const float* __restrict__ hw1_k2, const float* __restrict__ hw1_b2,
                           const float* __restrict__ b1_k,   const float* __restrict__ b1_b,
                           const float* __restrict__ hw2_k1, const float* __restrict__ hw2_b1,
                           const float* __restrict__ hw2_k2, const float* __restrict__ hw2_b2,
                           const float* __restrict__ hb2_k1, const float* __restrict__ hb2_b1,
                           const float* __restrict__ hb2_k2, const float* __restrict__ hb2_b2,
                           float* __restrict__ out) {
    __shared__ float Slds[ROWS * SLDA];                  // states tile 16x128 (stride 132)
    __shared__ float Hlds[ROWS * SLDA];                  // hidden (h1 / h2 / h3, reused)
    __shared__ float W1lds[ROWS * SLDA];                 // w1 (128 per row: [e]=w1a, [64+e]=w1b)
    __shared__ float B1lds[ROWS * SLDB];                 // b1 (64 per row)
    __shared__ float W2lds[ROWS * SLDB];                 // w2 (64 per row)
    __shared__ float B2lds[ROWS];                        // b2 scalar per row
    __shared__ float Qlds[ROWS * N_AGENTS];              // agent q per row
    __shared__ float CNlds[ROWS * S_SAMP * N_AGENTS];    // coalition means

    const int tid  = threadIdx.x;
    const int wave = tid >> 5;
    const int lane = tid & 31;
    const int row0 = blockIdx.x * ROWS;

#if USE_TDM
    // ---- Stage 0a: issue TDM DMA of the 16x128 f32 states tile into LDS. ----
    // 2D descriptor: dim0=128 elems (4B), dim1=8192 rows, stride0=128; tile 128x16 starting
    // at row row0. LDS padding done by the TDM itself: pad_interval=6 (every 128 DWORDs),
    // pad_amount=3 (+4 DWORDs) -> hardware-produced stride of 132 floats == SLDA.
    if (wave == 0) {
        const unsigned lds_addr = (unsigned)(size_t)(void*)Slds;   // low 32 bits = LDS offset
        const unsigned long long ga =
            (unsigned long long)(const void*)(states + (size_t)row0 * D_STATE);
        v4u g0;
        g0.x = 1u;                                              // count=1, user descriptor
        g0.y = lds_addr;                                        // lds_addr
        g0.z = (unsigned)(ga & 0xFFFFFFFFull);                  // global_addr[31:0]
        g0.w = (unsigned)((ga >> 32) & 0x01FFFFFFull)           // global_addr[56:32]
             | 0x80000000u;                                     // type=2 ("image")
        v8i g1;
        g1[0] = (int)0x07920000u;  // data_size=4B | pad_enable | pad_interval=6 | pad_amount=3
        g1[1] = (int)(128u << 16); // tensor_dim0 = 128
        g1[2] = (int)((unsigned)TB << 16);  // tensor_dim1 = 8192
        g1[3] = (int)(128u << 16); // tile_dim0 = 128
        g1[4] = (int)ROWS;         // tile_dim1 = 16
        g1[5] = 128;               // tensor_dim0_stride = 128
        g1[6] = 0;
        g1[7] = 0;
        v4i z4 = {0, 0, 0, 0};     // groups 2/3 unused (<=2D tensor)
        v8i z8 = {0, 0, 0, 0, 0, 0, 0, 0};
        __builtin_amdgcn_tensor_load_to_lds(g0, g1, z4, z4, z8, 0);
    }
#endif

    // ---- Stage 0b: agent q values (overlaps with TDM transfer) ----
    if (tid < ROWS * N_AGENTS) {
        const int r = tid >> 3, a = tid & 7;
        Qlds[tid] = q_vals[a * TB + row0 + r];          // q_vals is (n, t, b), row = t*B + b
    }

#if USE_TDM
    if (wave == 0) {
        __builtin_amdgcn_s_wait_tensorcnt(0);           // TENSORcnt is per issuing wave
    }
#else
    // Fallback: coalesced VGPR path for the states tile.
#pragma unroll
    for (int i = 0; i < ROWS; ++i)
        Slds[i * SLDA + tid] = states[(row0 + i) * D_STATE + tid];
#endif
    __syncthreads();

    // ---- Hypernet GEMM chain (each wave owns 16-column tiles) ----
    // h1 = relu(S @ hw1_k1 + hw1_b1)        (16 x 128)
    for (int c0 = wave * 16; c0 < HH_DIM; c0 += 64) {
        v8f acc = wmma_gemm_tile(Slds, SLDA, hw1_k1, HH_DIM, c0, D_STATE, lane);
        store_tile(Hlds, SLDA, c0, lane, acc, hw1_b1, 1);
    }
    __syncthreads();
    // w1 = abs(h1 @ hw1_k2 + hw1_b2)        (16 x 128)
    for (int c0 = wave * 16; c0 < 2 * E_DIM; c0 += 64) {
        v8f acc = wmma_gemm_tile(Hlds, SLDA, hw1_k2, 2 * E_DIM, c0, HH_DIM, lane);
        store_tile(W1lds, SLDA, c0, lane, acc, hw1_b2, 2);
    }
    // b1 = S @ b1_k + b1_b                  (16 x 64, one tile per wave)
    {
        const int c0 = wave * 16;
        v8f acc = wmma_gemm_tile(Slds, SLDA, b1_k, E_DIM, c0, D_STATE, lane);
        store_tile(B1lds, SLDB, c0, lane, acc, b1_b, 0);
    }
    __syncthreads();
    // h2 = relu(S @ hw2_k1 + hw2_b1)        (16 x 128, reuse Hlds)
    for (int c0 = wave * 16; c0 < HH_DIM; c0 += 64) {
        v8f acc = wmma_gemm_tile(Slds, SLDA, hw2_k1, HH_DIM, c0, D_STATE, lane);
        store_tile(Hlds, SLDA, c0, lane, acc, hw2_b1, 1);
    }
    __syncthreads();
    // w2 = abs(h2 @ hw2_k2 + hw2_b2)        (16 x 64)
    {
        const int c0 = wave * 16;
        v8f acc = wmma_gemm_tile(Hlds, SLDA, hw2_k2, E_DIM, c0, HH_DIM, lane);
        store_tile(W2lds, SLDB, c0, lane, acc, hw2_b2, 2);
    }
    __syncthreads();
    // h3 = relu(S @ hb2_k1 + hb2_b1)        (16 x 64, reuse Hlds)
    {
        const int c0 = wave * 16;
        v8f acc = wmma_gemm_tile(Slds, SLDA, hb2_k1, E_DIM, c0, D_STATE, lane);
        store_tile(Hlds, SLDA, c0, lane, acc, hb2_b1, 1);
    }
    __syncthreads();

    // ---- b2 scalar per row: h3 . hb2_k2 + hb2_b2 ----
    if (tid < ROWS) {
        float acc = hb2_b2[0];
#pragma unroll 8
        for (int e = 0; e < E_DIM; ++e)
            acc += Hlds[tid * SLDA + e] * hb2_k2[e];
        B2lds[tid] = acc;
    }

    // ---- Random coalition orderings: mean of predecessors' q per (row, sample, agent) ----
    for (int pair = tid; pair < ROWS * S_SAMP; pair += 128) {
        const int row_l = pair >> 4;
        const int s     = pair & 15;
        unsigned perm = 0x76543210u;   // nibble p = agent at position p
        unsigned rng  = pcg_hash((unsigned)((row0 + row_l) * S_SAMP + s) ^ 0x9E3779B9u);
        for (int i = 7; i > 0; --i) {  // Fisher-Yates on packed nibbles (no scratch)
            rng = pcg_hash(rng);
            const int j  = (int)(rng % (unsigned)(i + 1));
            const unsigned ni = (perm >> (4 * i)) & 0xFu;
            const unsigned nj = (perm >> (4 * j)) & 0xFu;
            perm &= ~((0xFu << (4 * i)) | (0xFu << (4 * j)));
            perm |= (nj << (4 * i)) | (ni << (4 * j));
        }
        float run = 0.0f;
        for (int p = 0; p < 8; ++p) {
            const int a = (int)((perm >> (4 * p)) & 0xFu);
            CNlds[(row_l * S_SAMP + s) * N_AGENTS + a] = (p > 0) ? run / (float)p : 0.0f;
            run += Qlds[row_l * N_AGENTS + a];
        }
    }
    __syncthreads();

    // ---- Fused ELU mixer: thread = (row_l, agent); loop samples; accumulate |y| ----
    {
        const int row_l = tid >> 3;
        const int a     = tid & 7;
        const float qv  = Qlds[row_l * N_AGENTS + a];
        const float b2v = B2lds[row_l];
        const float* w1a = &W1lds[row_l * SLDA];          // coal_norm weights
        const float* w1b = &W1lds[row_l * SLDA + E_DIM];  // q weights
        const float* b1r = &B1lds[row_l * SLDB];
        const float* w2r = &W2lds[row_l * SLDB];
        float sumabs = 0.0f;
        for (int s = 0; s < S_SAMP; ++s) {
            const float cn = CNlds[(row_l * S_SAMP + s) * N_AGENTS + a];
            float y = b2v;
#pragma unroll 8
            for (int e = 0; e < E_DIM; ++e) {
                const float x = cn * w1a[e] + qv * w1b[e] + b1r[e];
                const float h = (x > 0.0f) ? x : (__expf(x) - 1.0f);   // elu
                y += h * w2r[e];
            }
            sumabs += fabsf(y);
        }
        out[a * TB + row0 + row_l] = sumabs * (1.0f / (float)S_SAMP);
    }
}

extern "C" void kernel_launch(void* const* d_in, const int* in_sizes, int n_in,
                              void* d_out, int out_size, void* d_ws, size_t ws_size,
                              hipStream_t stream) {
    (void)in_sizes; (void)n_in; (void)out_size; (void)d_ws; (void)ws_size;
    const float* q_vals = (const float*)d_in[0];
    const float* states = (const float*)d_in[1];
    const float* hw1_k1 = (const float*)d_in[2];
    const float* hw1_b1 = (const float*)d_in[3];
    const float* hw1_k2 = (const float*)d_in[4];
    const float* hw1_b2 = (const float*)d_in[5];
    const float* b1_k   = (const float*)d_in[6];
    const float* b1_b   = (const float*)d_in[7];
    const float* hw2_k1 = (const float*)d_in[8];
    const float* hw2_b1 = (const float*)d_in[9];
    const float* hw2_k2 = (const float*)d_in[10];
    const float* hw2_b2 = (const float*)d_in[11];
    const float* hb2_k1 = (const float*)d_in[12];
    const float* hb2_b1 = (const float*)d_in[13];
    const float* hb2_k2 = (const float*)d_in[14];
    const float* hb2_b2 = (const float*)d_in[15];
    float* out = (float*)d_out;

    dim3 grid(TB / ROWS);   // 512 workgroups, 16 tb-rows each
    dim3 block(128);        // 4 waves (wave32)
    alpha_estimate_kernel<<<grid, block, 0, stream>>>(
        q_vals, states, hw1_k1, hw1_b1, hw1_k2, hw1_b2, b1_k, b1_b,
        hw2_k1, hw2_b1, hw2_k2, hw2_b2, hb2_k1, hb2_b1, hb2_k2, hb2_b2, out);
}